// HighDimFuzzyAttention_91233695302186
// MI455X (gfx1250) — compile-verified
//
#include <hip/hip_runtime.h>
#include <hip/hip_bf16.h>

#define B_  4
#define S_  2048
#define D_  8192
#define KSPLIT 8
#define DCHUNK 16              // columns per fused workgroup
#define NSLICE 16              // row slices per workgroup (256 threads / 16 cols)
#define ROWS_PER_SLICE (S_ / NSLICE)   // 128

typedef float v2f __attribute__((ext_vector_type(2)));
typedef float v4f __attribute__((ext_vector_type(4)));
typedef float v8f __attribute__((ext_vector_type(8)));

// ---------------------------------------------------------------------------
// y_c for one scalar element.
// mu_i = sigmoid(-(a_i*(x+b_i))) = 1/(1+exp(a_i*(x+b_i))); mu2 scaled by alpha.
// ---------------------------------------------------------------------------
__device__ __forceinline__ float yc_val(float xv, float kv, float mv,
                                        float a0, float a1, float a2,
                                        float b0, float b1, float b2, float alpha) {
    float mu0 = 1.0f / (1.0f + __expf(a0 * (xv + b0)));
    float mu1 = 1.0f / (1.0f + __expf(a1 * (xv + b1)));
    float mu2 = alpha / (1.0f + __expf(a2 * (xv + b2)));
    float w0 = 2.0f * fabsf(mu0 - 0.5f);
    float w1 = 2.0f * fabsf(mu1 - 0.5f);
    float w2 = 2.0f * fabsf(mu2 - 0.5f);
    float wc = fmaxf(w0, fmaxf(w1, w2));
    float Ic = 1.0f / (1.0f + __expf(kv * (mv - xv)));
    return Ic + wc - Ic * wc;
}

// ---------------------------------------------------------------------------
// Fused kernel: one workgroup owns slab x[b, :, d0..d0+15] (S*16 floats =
// 128KB), stages it into LDS once, then computes the column means AND the
// y_c reductions from LDS.  Saves one full HBM pass over x.
//   dynamic LDS layout:
//     xbuf   : S_*DCHUNK floats          (131072 B)
//     part_m : NSLICE*DCHUNK floats
//     part_n : NSLICE*DCHUNK floats
//     part_d : NSLICE*DCHUNK floats
//     mcol   : DCHUNK floats
// ---------------------------------------------------------------------------
__global__ __launch_bounds__(256) void fused_oc_kernel(const float* __restrict__ x,
                                                       const float* __restrict__ kparam,
                                                       const float* __restrict__ a3,
                                                       const float* __restrict__ b3,
                                                       const int* __restrict__ step_before,
                                                       const int* __restrict__ max_steps,
                                                       float* __restrict__ Oc) {
    extern __shared__ float lds[];
    float* xbuf   = lds;                              // S_*DCHUNK
    float* part_m = xbuf   + S_ * DCHUNK;             // NSLICE*DCHUNK
    float* part_n = part_m + NSLICE * DCHUNK;
    float* part_d = part_n + NSLICE * DCHUNK;
    float* mcol   = part_d + NSLICE * DCHUNK;

    const int t  = threadIdx.x;                       // 0..255
    const int wg = blockIdx.x;                        // [0, B * D/DCHUNK)
    const int b  = wg / (D_ / DCHUNK);
    const int d0 = (wg % (D_ / DCHUNK)) * DCHUNK;

    // ---- stage x slab into LDS (non-temporal: single-use stream) ----------
    {
        const v4f* __restrict__ gx =
            reinterpret_cast<const v4f*>(x + (size_t)b * S_ * D_ + d0);
        v4f* lds4 = reinterpret_cast<v4f*>(xbuf);
        // S_*DCHUNK/4 = 8192 float4s, 32 per thread
#pragma unroll 4
        for (int i = 0; i < (S_ * DCHUNK / 4) / 256; ++i) {
            const int f = t + 256 * i;                // float4 id
            const int s = f >> 2;                     // row
            const int q = f & 3;                      // quad within row
            v4f v = __builtin_nontemporal_load(&gx[(size_t)s * (D_ / 4) + q]);
            lds4[f] = v;
        }
    }
    __syncthreads();

    const int j     = t & (DCHUNK - 1);               // column 0..15
    const int slice = t >> 4;                         // 0..15

    // ---- column means from LDS -------------------------------------------
    {
        float acc = 0.f;
#pragma unroll 8
        for (int i = 0; i < ROWS_PER_SLICE; ++i) {
            const int row = slice + i * NSLICE;       // interleaved: conflict-free
            acc += xbuf[row * DCHUNK + j];
        }
        part_m[slice * DCHUNK + j] = acc;
    }
    __syncthreads();
    if (t < DCHUNK) {
        float s = 0.f;
#pragma unroll
        for (int sl = 0; sl < NSLICE; ++sl) s += part_m[sl * DCHUNK + t];
        mcol[t] = s * (1.0f / (float)S_);
    }
    __syncthreads();

    // ---- y_c reduction from LDS ------------------------------------------
    const float alpha = fminf(fmaxf((float)step_before[0] / (float)max_steps[0], 0.f), 1.f);
    const float a0 = a3[0], a1 = a3[1], a2 = a3[2];
    const float b0 = b3[0], b1 = b3[1], b2 = b3[2];
    const float kv = kparam[d0 + j];
    const float mv = mcol[j];

    {
        float num = 0.f, den = 0.f;
#pragma unroll 4
        for (int i = 0; i < ROWS_PER_SLICE; ++i) {
            const int row = slice + i * NSLICE;
            const float xv = xbuf[row * DCHUNK + j];
            const float y  = yc_val(xv, kv, mv, a0, a1, a2, b0, b1, b2, alpha);
            num += y * xv;
            den += y;
        }
        part_n[slice * DCHUNK + j] = num;
        part_d[slice * DCHUNK + j] = den;
    }
    __syncthreads();
    if (t < DCHUNK) {
        float num = 0.f, den = 0.f;
#pragma unroll
        for (int sl = 0; sl < NSLICE; ++sl) {
            num += part_n[sl * DCHUNK + t];
            den += part_d[sl * DCHUNK + t];
        }
        Oc[(size_t)b * D_ + d0 + t] = num / (den + 1e-6f);
    }
}

// ---------------------------------------------------------------------------
// Zero-pad Oc rows 4..15 so the WMMA A-fragment loads are unconditional.
// ---------------------------------------------------------------------------
__global__ __launch_bounds__(256) void pad_oc_kernel(float* __restrict__ Oc) {
    const int t = blockIdx.x * blockDim.x + threadIdx.x;   // [0, 12*D/4)
    v4f z = {0.f, 0.f, 0.f, 0.f};
    reinterpret_cast<v4f*>(Oc + (size_t)B_ * D_)[t] = z;
}

// ---------------------------------------------------------------------------
// Skinny GEMM via V_WMMA_F32_16X16X4_F32.
// proj_partial[ks, m, e] = sum_{d in chunk ks} Oc[m,d] * W[e,d]
// A = Oc (16 x K), B = W^T (B[k,n] = W[n*D+k]).  One wave per 16x16 N-tile
// per K-chunk.  A-fragment (16x4 f32): lanes 0-15 hold K=0,1; lanes 16-31
// hold K=2,3.  B mirrored.  D-fragment: VGPR r = row M=r on lanes 0-15.
// ---------------------------------------------------------------------------
__global__ __launch_bounds__(32) void proj_wmma_kernel(const float* __restrict__ Oc,
                                                       const float* __restrict__ W,
                                                       float* __restrict__ partial) {
    const int lane  = threadIdx.x;            // 0..31
    const int n0    = blockIdx.x * 16;        // output-column tile
    const int ks    = blockIdx.y;             // K chunk id
    const int kbeg  = ks * (D_ / KSPLIT);
    const int kend  = kbeg + (D_ / KSPLIT);
    const int mrow  = lane & 15;              // A row / B column within tile
    const int khalf = (lane >> 4) * 2;        // 0 for lanes 0-15, 2 for 16-31

    const float* __restrict__ arow = Oc + (size_t)mrow * D_;
    const float* __restrict__ brow = W  + (size_t)(n0 + mrow) * D_;

    v8f c = {};
#pragma unroll 4
    for (int k = kbeg; k < kend; k += 4) {
        const v2f av = *reinterpret_cast<const v2f*>(arow + k + khalf);
        const v2f bv = *reinterpret_cast<const v2f*>(brow + k + khalf);
        c = __builtin_amdgcn_wmma_f32_16x16x4_f32(false, av, false, bv,
                                                  (short)0, c, false, false);
    }
    if (lane < 16) {
        float* p = partial + (size_t)ks * 4 * D_ + n0 + lane;
        p[0 * (size_t)D_] = c[0];
        p[1 * (size_t)D_] = c[1];
        p[2 * (size_t)D_] = c[2];
        p[3 * (size_t)D_] = c[3];
    }
}

// ---------------------------------------------------------------------------
// Deterministic reduction of K-split partials + bias.
// ---------------------------------------------------------------------------
__global__ __launch_bounds__(256) void reduce_proj_kernel(const float* __restrict__ partial,
                                                          const float* __restrict__ bproj,
                                                          float* __restrict__ proj) {
    const int t = blockIdx.x * blockDim.x + threadIdx.x;    // [0, 4*D)
    const int e = t % D_;
    float s = bproj[e];
#pragma unroll
    for (int ks = 0; ks < KSPLIT; ++ks)
        s += partial[(size_t)ks * 4 * D_ + t];
    proj[t] = s;
}

// ---------------------------------------------------------------------------
// out[b,s,d] = x[b,s,d] * proj[b,d].  float4, non-temporal streams.
// ---------------------------------------------------------------------------
__global__ __launch_bounds__(256) void scale_out_kernel(const float* __restrict__ x,
                                                        const float* __restrict__ proj,
                                                        float* __restrict__ out) {
    const size_t t = (size_t)blockIdx.x * blockDim.x + threadIdx.x;  // [0, B*S*D/4)
    const v4f xv = __builtin_nontemporal_load(&reinterpret_cast<const v4f*>(x)[t]);
    const size_t elem = t * 4;
    const int b = (int)(elem / ((size_t)S_ * D_));
    const int d = (int)(elem & (size_t)(D_ - 1));
    const v4f pv = reinterpret_cast<const v4f*>(proj)[((size_t)b * D_ + d) >> 2];
    const v4f o = xv * pv;
    __builtin_nontemporal_store(o, &reinterpret_cast<v4f*>(out)[t]);
}

// ---------------------------------------------------------------------------
extern "C" void kernel_launch(void* const* d_in, const int* in_sizes, int n_in,
                              void* d_out, int out_size, void* d_ws, size_t ws_size,
                              hipStream_t stream) {
    const float* x      = (const float*)d_in[0];
    const float* kparam = (const float*)d_in[1];
    const float* a3     = (const float*)d_in[2];
    const float* b3     = (const float*)d_in[3];
    const float* W      = (const float*)d_in[4];
    const float* bproj  = (const float*)d_in[5];
    const int*   sb     = (const int*)d_in[6];
    const int*   ms     = (const int*)d_in[7];
    float*       out    = (float*)d_out;

    // workspace layout
    char* ws = (char*)d_ws;
    float* Oc_buf  = (float*)(ws);                               // 16*D (rows 4..15 zero)
    float* partial = (float*)(ws + (size_t)16 * D_ * 4);         // KSPLIT*4*D
    float* proj    = (float*)(ws + (size_t)16 * D_ * 4
                                 + (size_t)KSPLIT * 4 * D_ * 4); // 4*D

    // 1) fused mean + O_c via 128KB LDS slab per workgroup
    const size_t lds_bytes = ((size_t)S_ * DCHUNK + 3 * NSLICE * DCHUNK + DCHUNK) * sizeof(float);
    fused_oc_kernel<<<B_ * (D_ / DCHUNK), 256, lds_bytes, stream>>>(
        x, kparam, a3, b3, sb, ms, Oc_buf);

    // 1b) zero-pad Oc rows 4..15
    pad_oc_kernel<<<(12 * D_ / 4) / 256, 256, 0, stream>>>(Oc_buf);

    // 2) WMMA GEMM, K-split partials
    dim3 ggrid(D_ / 16, KSPLIT);
    proj_wmma_kernel<<<ggrid, 32, 0, stream>>>(Oc_buf, W, partial);

    // 3) reduce partials + bias
    reduce_proj_kernel<<<(4 * D_) / 256, 256, 0, stream>>>(partial, bproj, proj);

    // 4) out = x * proj
    scale_out_kernel<<<(int)(((size_t)B_ * S_ * D_ / 4) / 256), 256, 0, stream>>>(x, proj, out);
}